// m_GCN_35716948034104
// MI455X (gfx1250) — compile-verified
//
#include <hip/hip_runtime.h>
#include <math.h>

typedef __attribute__((ext_vector_type(2))) float v2f;
typedef __attribute__((ext_vector_type(8))) float v8f;

#define GCN_EPS 1e-7f
#define STG 68   // staging row stride (floats): breaks LDS bank conflicts, keeps 8B align

__device__ __forceinline__ float selu_f(float x) {
    const float alpha = 1.6732632423543772f;
    const float scale = 1.0507009873554805f;
    return x > 0.0f ? scale * x : scale * alpha * (__expf(x) - 1.0f);
}

// ---------------------------------------------------------------------------
// init: Z = x * Win (outer product) ; agg = 0
// ---------------------------------------------------------------------------
__global__ void gcn_init(const float* __restrict__ x, const float* __restrict__ Win,
                         float* __restrict__ Z, float* __restrict__ agg, int N) {
    int t = blockIdx.x * blockDim.x + threadIdx.x;
    if (t < N * 64) agg[t] = 0.0f;
    if (t < N * 32) Z[t] = x[t >> 5] * Win[t & 31];
}

// ---------------------------------------------------------------------------
// edge phase: one wave per edge, lane = latent dim (0..31).
// msg = selu(concat[sndr, ea0 + |sndr-rcvr|]) + eps, scatter-add into agg[dst].
// Gathers/atomics are 128B coalesced per wave; Z & agg live in 192MB L2.
// ---------------------------------------------------------------------------
__global__ void __launch_bounds__(256) gcn_edge(
        const float* __restrict__ Z, const int* __restrict__ ei,
        const float* __restrict__ eattr, const float* __restrict__ Wedge,
        float* agg, int E) {
    int gt    = blockIdx.x * blockDim.x + threadIdx.x;
    int wave  = gt >> 5;
    int lane  = gt & 31;
    int nwave = (gridDim.x * blockDim.x) >> 5;
    float we0 = Wedge[lane];        // Wedge[0][lane]
    float we1 = Wedge[32 + lane];   // Wedge[1][lane]
    for (int e = wave; e < E; e += nwave) {
        int s = ei[e];              // wave-uniform
        int d = ei[E + e];
        float a0 = eattr[2 * e];
        float a1 = eattr[2 * e + 1];
        float sv = Z[(size_t)s * 32 + lane];
        float rv = Z[(size_t)d * 32 + lane];
        float ea = we0 * a0 + we1 * a1 + fabsf(sv - rv);
        float mlo = selu_f(sv) + GCN_EPS;
        float mhi = selu_f(ea) + GCN_EPS;
        unsafeAtomicAdd(&agg[(size_t)d * 64 + lane],       mlo);
        unsafeAtomicAdd(&agg[(size_t)d * 64 + 32 + lane],  mhi);
    }
}

// ---------------------------------------------------------------------------
// node phase: per wave, a 16-node tile.
//   G1: lat = agg[16x64] @ wres[64x32] + Z      (residual preloaded into C)
//   G2: h   = selu(lat[16x32] @ w1[32x64])
//   G3: Zn  = selu(h[16x64] @ w2[64x32])
// All GEMMs via V_WMMA_F32_16X16X4_F32. Weights staged transposed in LDS,
// intermediates staged through per-wave LDS tile (C-layout -> A-layout).
// Also zeroes its agg rows for the next layer.
// ---------------------------------------------------------------------------
__global__ void __launch_bounds__(256) gcn_node(
        float* agg,                       // read then zeroed: NOT restrict (ordering!)
        const float* __restrict__ Zin,
        float* __restrict__ Zout,
        const float* __restrict__ wres,   // [64][32] this layer
        const float* __restrict__ w1,     // [32][64]
        const float* __restrict__ w2,     // [64][32]
        int ntiles) {
    __shared__ float sWres[32 * 64];      // transposed: [n][k]
    __shared__ float sW1  [64 * 32];      // transposed: [n][k]
    __shared__ float sW2  [32 * 64];      // transposed: [n][k]
    __shared__ float sStage[8][16 * STG]; // per-wave [16 x <=64] staging

    int tid = threadIdx.x;
    for (int i = tid; i < 2048; i += 256) {
        int r = i >> 5, c = i & 31;        // wres/w2: [64][32]
        sWres[c * 64 + r] = wres[i];
        sW2  [c * 64 + r] = w2[i];
        int r1 = i >> 6, c1 = i & 63;      // w1: [32][64]
        sW1[c1 * 32 + r1] = w1[i];
    }
    __syncthreads();

    int wave = tid >> 5, lane = tid & 31;
    int tile = blockIdx.x * 8 + wave;      // wave-uniform -> EXEC stays all-ones
    if (tile >= ntiles) return;
    int m0   = tile << 4;
    int row  = lane & 15;                  // M for A/C, N for B
    int half = lane >> 4;
    float* stage = sStage[wave];

    // ---- G1: C preloaded with residual Z, then 16 K-steps x 2 N-tiles ----
    v8f c0, c1;
#pragma unroll
    for (int v = 0; v < 8; ++v) {
        size_t zr = (size_t)(m0 + v + 8 * half) * 32;
        c0[v] = Zin[zr + row];
        c1[v] = Zin[zr + 16 + row];
    }
    const float* aRow = agg + (size_t)(m0 + row) * 64;
#pragma unroll
    for (int k = 0; k < 16; ++k) {
        v2f a  = *(const v2f*)(aRow + 4 * k + 2 * half);
        v2f b0 = *(const v2f*)(sWres + (size_t)row        * 64 + 4 * k + 2 * half);
        v2f b1 = *(const v2f*)(sWres + (size_t)(16 + row) * 64 + 4 * k + 2 * half);
        c0 = __builtin_amdgcn_wmma_f32_16x16x4_f32(false, a, false, b0, (short)0, c0, false, false);
        c1 = __builtin_amdgcn_wmma_f32_16x16x4_f32(false, a, false, b1, (short)0, c1, false, false);
    }
#pragma unroll
    for (int v = 0; v < 8; ++v) {          // stage lat [16x32]
        int r2 = (v + 8 * half) * STG;
        stage[r2 + row]      = c0[v];
        stage[r2 + 16 + row] = c1[v];
    }

    // ---- G2: h = selu(lat @ w1) : 8 K-steps x 4 N-tiles ----
    v8f d0 = {}, d1 = {}, d2 = {}, d3 = {};
#pragma unroll
    for (int k = 0; k < 8; ++k) {
        v2f a  = *(const v2f*)(stage + row * STG + 4 * k + 2 * half);
        v2f b0 = *(const v2f*)(sW1 + (size_t)row        * 32 + 4 * k + 2 * half);
        v2f b1 = *(const v2f*)(sW1 + (size_t)(16 + row) * 32 + 4 * k + 2 * half);
        v2f b2 = *(const v2f*)(sW1 + (size_t)(32 + row) * 32 + 4 * k + 2 * half);
        v2f b3 = *(const v2f*)(sW1 + (size_t)(48 + row) * 32 + 4 * k + 2 * half);
        d0 = __builtin_amdgcn_wmma_f32_16x16x4_f32(false, a, false, b0, (short)0, d0, false, false);
        d1 = __builtin_amdgcn_wmma_f32_16x16x4_f32(false, a, false, b1, (short)0, d1, false, false);
        d2 = __builtin_amdgcn_wmma_f32_16x16x4_f32(false, a, false, b2, (short)0, d2, false, false);
        d3 = __builtin_amdgcn_wmma_f32_16x16x4_f32(false, a, false, b3, (short)0, d3, false, false);
    }
#pragma unroll
    for (int v = 0; v < 8; ++v) {          // stage h [16x64] (overwrites lat; DS in-order)
        int r2 = (v + 8 * half) * STG;
        stage[r2 + row]      = selu_f(d0[v]);
        stage[r2 + 16 + row] = selu_f(d1[v]);
        stage[r2 + 32 + row] = selu_f(d2[v]);
        stage[r2 + 48 + row] = selu_f(d3[v]);
    }

    // ---- G3: Zn = selu(h @ w2) : 16 K-steps x 2 N-tiles ----
    v8f e0 = {}, e1 = {};
#pragma unroll
    for (int k = 0; k < 16; ++k) {
        v2f a  = *(const v2f*)(stage + row * STG + 4 * k + 2 * half);
        v2f b0 = *(const v2f*)(sW2 + (size_t)row        * 64 + 4 * k + 2 * half);
        v2f b1 = *(const v2f*)(sW2 + (size_t)(16 + row) * 64 + 4 * k + 2 * half);
        e0 = __builtin_amdgcn_wmma_f32_16x16x4_f32(false, a, false, b0, (short)0, e0, false, false);
        e1 = __builtin_amdgcn_wmma_f32_16x16x4_f32(false, a, false, b1, (short)0, e1, false, false);
    }
#pragma unroll
    for (int v = 0; v < 8; ++v) {
        size_t zr = (size_t)(m0 + v + 8 * half) * 32;
        Zout[zr + row]      = selu_f(e0[v]);
        Zout[zr + 16 + row] = selu_f(e1[v]);
    }

    // zero this tile's agg rows for the next layer (lane -> (row, half-row))
    float4 z4 = make_float4(0.f, 0.f, 0.f, 0.f);
    float* ar = agg + (size_t)(m0 + (lane >> 1)) * 64 + (lane & 1) * 32;
#pragma unroll
    for (int i = 0; i < 8; ++i) ((float4*)ar)[i] = z4;
}

// ---------------------------------------------------------------------------
// output: sigmoid(Z @ Wout), one thread per node (32-float contiguous row)
// ---------------------------------------------------------------------------
__global__ void gcn_out(const float* __restrict__ Z, const float* __restrict__ Wout,
                        float* __restrict__ out, int N) {
    int t = blockIdx.x * blockDim.x + threadIdx.x;
    if (t >= N) return;
    const float4* z = (const float4*)(Z + (size_t)t * 32);
    const float4* w = (const float4*)Wout;
    float acc = 0.0f;
#pragma unroll
    for (int i = 0; i < 8; ++i) {
        float4 zv = z[i], wv = w[i];
        acc += zv.x * wv.x + zv.y * wv.y + zv.z * wv.z + zv.w * wv.w;
    }
    out[t] = 1.0f / (1.0f + __expf(-acc));
}

// ---------------------------------------------------------------------------
extern "C" void kernel_launch(void* const* d_in, const int* in_sizes, int n_in,
                              void* d_out, int out_size, void* d_ws, size_t ws_size,
                              hipStream_t stream) {
    const float* x     = (const float*)d_in[0];
    const int*   ei    = (const int*)  d_in[1];
    const float* eattr = (const float*)d_in[2];
    const float* Win   = (const float*)d_in[3];
    const float* Wedge = (const float*)d_in[4];
    const float* Wres  = (const float*)d_in[5];
    const float* W1    = (const float*)d_in[6];
    const float* W2    = (const float*)d_in[7];
    const float* Wout  = (const float*)d_in[8];
    float* out = (float*)d_out;

    int N = in_sizes[0];              // 100000
    int E = in_sizes[1] / 2;          // 3200000
    int L = in_sizes[5] / (64 * 32);  // 45

    // workspace layout: Za [N*32] | Zb [N*32] | agg [N*64]  (51.2 MB total)
    float* Za  = (float*)d_ws;
    float* Zb  = Za + (size_t)N * 32;
    float* agg = Zb + (size_t)N * 32;

    {
        int total  = N * 64;
        int blocks = (total + 255) / 256;
        gcn_init<<<blocks, 256, 0, stream>>>(x, Win, Za, agg, N);
    }

    int ntiles     = N / 16;          // N=100000 is a multiple of 16
    int nodeBlocks = (ntiles + 7) / 8;
    int edgeBlocks = 4096;            // 32768 waves, grid-stride over E

    float* Zc = Za;
    float* Zn = Zb;
    for (int l = 0; l < L; ++l) {
        gcn_edge<<<edgeBlocks, 256, 0, stream>>>(Zc, ei, eattr, Wedge, agg, E);
        gcn_node<<<nodeBlocks, 256, 0, stream>>>(agg, Zc, Zn,
            Wres + (size_t)l * 2048, W1 + (size_t)l * 2048, W2 + (size_t)l * 2048,
            ntiles);
        float* t = Zc; Zc = Zn; Zn = t;
    }

    {
        int blocks = (N + 255) / 256;
        gcn_out<<<blocks, 256, 0, stream>>>(Zc, Wout, out, N);
    }
}